// GraphConvolution_6201932775567
// MI455X (gfx1250) — compile-verified
//
#include <hip/hip_runtime.h>
#include <hip/hip_bf16.h>

// ---------------------------------------------------------------------------
// GraphConvolution on MI455X (gfx1250):
//   support = X @ W          (WMMA f32 16x16x4; X read exactly once; W staged
//                             into LDS via Tensor Data Mover, one DMA / block)
//   out     = segsum(support[col]*val, row) + bias   (L2-resident COO SpMM)
// ---------------------------------------------------------------------------

typedef float v2f  __attribute__((ext_vector_type(2)));
typedef float v8f  __attribute__((ext_vector_type(8)));
typedef unsigned int u32x4 __attribute__((ext_vector_type(4)));
typedef int   i32x4 __attribute__((ext_vector_type(4)));
typedef int   i32x8 __attribute__((ext_vector_type(8)));

#define IN_F  128
#define OUT_F 64
#define WAVES_PER_BLOCK 8
#define ROWS_PER_BLOCK  (WAVES_PER_BLOCK * 16)

// ---------------------------------------------------------------------------
// TDM: DMA the 128x64 f32 weight matrix (32 KB) from global into LDS.
// D# per CDNA5 ISA §8:
//   group0: count=1 | lds_addr | global_addr[56:0] | type=2
//   group1: data_size=2(4B) | tensor_dim0=64 | tensor_dim1=128 |
//           tile_dim0=64 | tile_dim1=128 | tensor_dim0_stride=64
//   groups 2/3 unused (<=2D tensor) -> zero-filled.
// ---------------------------------------------------------------------------
#if __has_builtin(__builtin_amdgcn_tensor_load_to_lds)
#define HAVE_TDM 1
__device__ __forceinline__ void tdm_stage_w(const float* w, unsigned lds_off) {
    unsigned long long ga = (unsigned long long)(uintptr_t)w;
    u32x4 g0;
    g0.x = 1u;                                               // count=1
    g0.y = lds_off;                                          // lds_addr (bytes)
    g0.z = (unsigned)(ga & 0xFFFFFFFFull);                   // global_addr[31:0]
    g0.w = (unsigned)((ga >> 32) & 0x01FFFFFFull) | (2u << 30); // [56:32] | type=2
    i32x8 g1;
    g1[0] = 0x00020000;        // workgroup_mask=0, data_size=2 (4 bytes)
    g1[1] = (OUT_F << 16);     // tensor_dim0 = 64   (bits 79:48, low half)
    g1[2] = (IN_F  << 16);     // tensor_dim1 = 128  (bits 111:80, low half)
    g1[3] = (OUT_F << 16);     // tile_dim0 = 64     (bits 127:112)
    g1[4] = IN_F;              // tile_dim1 = 128, tile_dim2 = 0
    g1[5] = OUT_F;             // tensor_dim0_stride = 64 (bits 207:160, low)
    g1[6] = 0;
    g1[7] = 0;
    i32x4 z4 = {0, 0, 0, 0};
#if __clang_major__ >= 23
    i32x8 z8 = {0, 0, 0, 0, 0, 0, 0, 0};
    __builtin_amdgcn_tensor_load_to_lds(g0, g1, z4, z4, z8, 0);
#else
    __builtin_amdgcn_tensor_load_to_lds(g0, g1, z4, z4, 0);
#endif
}
#else
#define HAVE_TDM 0
#endif

// ---------------------------------------------------------------------------
// Kernel 1: support[N,64] = X[N,128] @ W[128,64]
// Block = 256 threads = 8 wave32. Each wave owns one 16-row tile of X and
// computes the full 16x64 output stripe with 4 accumulators, so each A
// fragment feeds 4 WMMAs (X read exactly once device-wide). B fragments come
// from the LDS-staged W copy.
//
// f32 A (16x4): lanes 0-15 -> M=lane, K={k0,k0+1}; lanes 16-31 -> K={k0+2,k0+3}.
// f32 B (4x16): lanes 0-15 -> N=lane, K={k0,k0+1}; lanes 16-31 -> K={k0+2,k0+3}.
// f32 C/D     : VGPR r -> M=r (lanes 0-15) / M=r+8 (lanes 16-31), N=lane&15.
// ---------------------------------------------------------------------------
__global__ void gcn_gemm_wmma(const float* __restrict__ x,
                              const float* __restrict__ w,
                              float* __restrict__ support,
                              int n_nodes) {
    __shared__ float wlds[IN_F * OUT_F];   // 32 KB of the 320 KB WGP LDS

    const int lane = threadIdx.x & 31;
    const int wave = threadIdx.x >> 5;     // 0..7: row tile within block
    const int half = lane >> 4;            // 0: lanes 0-15, 1: lanes 16-31
    const int l    = lane & 15;

    // ---- Stage W into LDS (one TDM DMA per block) ----
#if HAVE_TDM
    if (wave == 0) {
        tdm_stage_w(w, (unsigned)(uintptr_t)&wlds[0]);
        __builtin_amdgcn_s_wait_tensorcnt(0);
    }
#else
    for (int i = threadIdx.x; i < (IN_F * OUT_F) / 4; i += blockDim.x)
        ((float4*)wlds)[i] = ((const float4*)w)[i];
#endif
    __syncthreads();

    // ---- Per-wave 16x64 GEMM stripe ----
    const int row0 = (blockIdx.x * WAVES_PER_BLOCK + wave) * 16;
    int arow = row0 + l;
    if (arow > n_nodes - 1) arow = n_nodes - 1;   // clamp reads, EXEC stays full
    const float* __restrict__ aptr = x + (size_t)arow * IN_F;

    v8f acc0 = {}, acc1 = {}, acc2 = {}, acc3 = {};
    #pragma unroll 4
    for (int k0 = 0; k0 < IN_F; k0 += 4) {
        const int ka = k0 + half * 2;      // this lane's K pair
        v2f a;
        a.x = aptr[ka];
        a.y = aptr[ka + 1];
        const float* wp0 = &wlds[ka * OUT_F + l];
        const float* wp1 = &wlds[(ka + 1) * OUT_F + l];
        v2f b0; b0.x = wp0[ 0]; b0.y = wp1[ 0];
        v2f b1; b1.x = wp0[16]; b1.y = wp1[16];
        v2f b2; b2.x = wp0[32]; b2.y = wp1[32];
        v2f b3; b3.x = wp0[48]; b3.y = wp1[48];
        acc0 = __builtin_amdgcn_wmma_f32_16x16x4_f32(false, a, false, b0, (short)0, acc0, false, false);
        acc1 = __builtin_amdgcn_wmma_f32_16x16x4_f32(false, a, false, b1, (short)0, acc1, false, false);
        acc2 = __builtin_amdgcn_wmma_f32_16x16x4_f32(false, a, false, b2, (short)0, acc2, false, false);
        acc3 = __builtin_amdgcn_wmma_f32_16x16x4_f32(false, a, false, b3, (short)0, acc3, false, false);
    }

    // ---- Store D tiles: VGPR r -> row (row0 + half*8 + r) ----
    const int mrow = row0 + half * 8;
    if (mrow + 7 < n_nodes) {
        float* __restrict__ outp = support + (size_t)mrow * OUT_F + l;
        #pragma unroll
        for (int r = 0; r < 8; ++r) {
            float* o = outp + (size_t)r * OUT_F;
            o[ 0] = acc0[r];
            o[16] = acc1[r];
            o[32] = acc2[r];
            o[48] = acc3[r];
        }
    } else {
        #pragma unroll
        for (int r = 0; r < 8; ++r) {
            if (mrow + r < n_nodes) {
                float* o = support + (size_t)(mrow + r) * OUT_F + l;
                o[ 0] = acc0[r];
                o[16] = acc1[r];
                o[32] = acc2[r];
                o[48] = acc3[r];
            }
        }
    }
}

// ---------------------------------------------------------------------------
// Kernel 2: out[i] = bias[i % 64]  (vectorized float4; pure-accumulate SpMM)
// ---------------------------------------------------------------------------
__global__ void gcn_init_bias(const float* __restrict__ bias,
                              float* __restrict__ out,
                              int total4) {
    int i = blockIdx.x * blockDim.x + threadIdx.x;
    if (i < total4)
        ((float4*)out)[i] = ((const float4*)bias)[i & (OUT_F / 4 - 1)];
}

// ---------------------------------------------------------------------------
// Kernel 3: COO SpMM scatter. One wave32 per edge: coalesced 256B float2
// gather from L2-resident support, 2x global_atomic_add_f32 per lane into
// L2-resident out. Edge metadata routed through readfirstlane -> scalar
// loads; edge stream prefetched (global_prefetch_b8).
// ---------------------------------------------------------------------------
__global__ void gcn_spmm_scatter(const float* __restrict__ support,
                                 const int*   __restrict__ erow,
                                 const int*   __restrict__ ecol,
                                 const float* __restrict__ evals,
                                 float* __restrict__ out,
                                 int n_edges) {
    const int e    = (blockIdx.x * blockDim.x + threadIdx.x) >> 5;
    const int lane = threadIdx.x & 31;
    if (e >= n_edges) return;

    const int eu = __builtin_amdgcn_readfirstlane(e);   // wave-uniform
    __builtin_prefetch(erow + eu + 2048, 0, 1);         // speculative, safe OOB
    __builtin_prefetch(ecol + eu + 2048, 0, 1);

    const int   r  = erow[eu];
    const int   c  = ecol[eu];
    const float ev = evals[eu];

    const float2 v = *((const float2*)(support + (size_t)c * OUT_F) + lane);
    float* dst = out + (size_t)r * OUT_F + lane * 2;
    atomicAdd(dst + 0, v.x * ev);
    atomicAdd(dst + 1, v.y * ev);
}

// ---------------------------------------------------------------------------
// Launcher. Inputs: x[N,128] f32, weight[128,64] f32, bias[64] f32,
//                   edge_row[E] i32, edge_col[E] i32, edge_vals[E] f32.
// d_ws: support[N,64] f32 (25.6 MB).
// ---------------------------------------------------------------------------
extern "C" void kernel_launch(void* const* d_in, const int* in_sizes, int n_in,
                              void* d_out, int out_size, void* d_ws, size_t ws_size,
                              hipStream_t stream) {
    const float* x     = (const float*)d_in[0];
    const float* wgt   = (const float*)d_in[1];
    const float* bias  = (const float*)d_in[2];
    const int*   erow  = (const int*)d_in[3];
    const int*   ecol  = (const int*)d_in[4];
    const float* evals = (const float*)d_in[5];
    float*       out   = (float*)d_out;
    float*       supp  = (float*)d_ws;

    const int n_nodes = in_sizes[0] / IN_F;
    const int n_edges = in_sizes[3];

    // 1) GEMM: 8 wave32 per block, each wave computes a 16x64 output stripe.
    {
        dim3 grid((n_nodes + ROWS_PER_BLOCK - 1) / ROWS_PER_BLOCK);
        gcn_gemm_wmma<<<grid, WAVES_PER_BLOCK * 32, 0, stream>>>(x, wgt, supp, n_nodes);
    }
    // 2) Seed output with bias (float4-wide).
    {
        const int total4 = n_nodes * OUT_F / 4;
        dim3 grid((total4 + 255) / 256);
        gcn_init_bias<<<grid, 256, 0, stream>>>(bias, out, total4);
    }
    // 3) Edge scatter: one wave per edge, 8 edges per 256-thread block.
    {
        long long threads = (long long)n_edges * 32;
        dim3 grid((unsigned)((threads + 255) / 256));
        gcn_spmm_scatter<<<grid, 256, 0, stream>>>(supp, erow, ecol, evals, out, n_edges);
    }
}